// AttnGate_75256416961051
// MI455X (gfx1250) — compile-verified
//
#include <hip/hip_runtime.h>
#include <hip/hip_bf16.h>

#define KH    8
#define G     4
#define D     128
#define DG    128
#define BLOCK 64
#define MAXS  4096
#define NB    64
#define BSZ   4

typedef __attribute__((ext_vector_type(2))) float v2f;
typedef __attribute__((ext_vector_type(8))) float v8f;

static __device__ __forceinline__ v8f zero8() {
    v8f z;
    #pragma unroll
    for (int i = 0; i < 8; ++i) z[i] = 0.0f;
    return z;
}

// ---------------------------------------------------------------------------
// Kernel 1: per-head Q projection (T x 512) * (512 x 128) with fp32 WMMA,
// fused rmsnorm + rope epilogue, scatter to padded (b,h,s,128) layout.
// Block = 256 threads (8 waves); tile M=128 (tokens) x N=128, K staged in 32s.
// A: [m][k] stride 36 -> aligned ds_load_b64 fragments, conflict-free.
// B: k-pair packed [kp][2n+parity] stride 160 -> aligned b64, conflict-free.
// ---------------------------------------------------------------------------
__global__ __launch_bounds__(256) void qproj_kernel(
    const float* __restrict__ q, const float* __restrict__ wq,
    const float* __restrict__ qnw, const float* __restrict__ cosq,
    const float* __restrict__ sinq, const long long* __restrict__ unpad,
    float* __restrict__ qrot, int T)
{
    __shared__ float As[128 * 36];
    __shared__ float Bs[16 * 160];

    const int tid = threadIdx.x, wave = tid >> 5, lane = tid & 31;
    const int h  = blockIdx.y;
    const int t0 = blockIdx.x * 128;

    v8f acc[8];
    #pragma unroll
    for (int i = 0; i < 8; ++i) acc[i] = zero8();

    const float* Ag = q + (size_t)t0 * (KH * G * D) + (size_t)h * (G * D); // row stride 4096
    const float* Bg = wq + (size_t)h * (G * D) * DG;                       // row stride 128

    for (int kc = 0; kc < G * D; kc += 32) {
        // stage A chunk: 128 rows x 32 k
        for (int i = tid; i < 128 * 8; i += 256) {
            int row = i >> 3, c4 = (i & 7) * 4;
            const float* src = Ag + (size_t)row * (KH * G * D) + kc + c4;
            float4 v = (t0 + row < T) ? *(const float4*)src
                                      : make_float4(0.f, 0.f, 0.f, 0.f);
            float* dst = &As[row * 36 + c4];
            dst[0] = v.x; dst[1] = v.y; dst[2] = v.z; dst[3] = v.w;
            if (kc + 32 < G * D) __builtin_prefetch(src + 32, 0, 0);
        }
        // stage B chunk: 32 k x 128 n, packed in k-pairs
        for (int i = tid; i < 32 * 32; i += 256) {
            int row = i >> 5, c4 = (i & 31) * 4;
            float4 v = *(const float4*)(Bg + (size_t)(kc + row) * DG + c4);
            float* dst = &Bs[(row >> 1) * 160 + (row & 1)];
            dst[(c4 + 0) * 2] = v.x; dst[(c4 + 1) * 2] = v.y;
            dst[(c4 + 2) * 2] = v.z; dst[(c4 + 3) * 2] = v.w;
        }
        __syncthreads();

        const int mrow = wave * 16 + (lane & 15);
        const int half = lane >> 4;               // lanes 16-31 hold K+2 of fragment
        for (int kk = 0; kk < 32; kk += 4) {
            int kb = kk + half * 2;
            v2f a = *(const v2f*)&As[mrow * 36 + kb];
            int kp = (kk >> 1) + half;
            #pragma unroll
            for (int nt = 0; nt < 8; ++nt) {
                int n = nt * 16 + (lane & 15);
                v2f b = *(const v2f*)&Bs[kp * 160 + 2 * n];
                acc[nt] = __builtin_amdgcn_wmma_f32_16x16x4_f32(
                    false, a, false, b, (short)0, acc[nt], false, false);
            }
        }
        __syncthreads();
    }

    // epilogue: rmsnorm + rope, all in-register
    const int g = lane >> 4;
    #pragma unroll
    for (int r = 0; r < 8; ++r) {
        int m = wave * 16 + r + g * 8;
        int t = t0 + m;
        if (t >= T) continue;
        long long up = unpad[t];
        int b = (int)(up / MAXS);
        int s = (int)(up % MAXS);
        float p = 0.f;
        #pragma unroll
        for (int nt = 0; nt < 8; ++nt) { float v = acc[nt][r]; p += v * v; }
        p += __shfl_xor(p, 1); p += __shfl_xor(p, 2);
        p += __shfl_xor(p, 4); p += __shfl_xor(p, 8);   // stays inside 16-lane half
        float rinv = 1.0f / sqrtf(p * (1.0f / DG) + 1e-6f);
        const float* cq = cosq + ((size_t)b * MAXS + s) * DG;
        const float* sq = sinq + ((size_t)b * MAXS + s) * DG;
        float* outp = qrot + (((size_t)b * KH + h) * MAXS + s) * DG;
        #pragma unroll
        for (int nt = 0; nt < 8; ++nt) {
            int o = nt * 16 + (lane & 15);
            float y  = acc[nt][r]     * rinv * qnw[o];
            float y2 = acc[nt ^ 4][r] * rinv * qnw[o ^ 64];  // rotate-half partner, same lane
            float rot = (nt < 4) ? -y2 : y2;
            outp[o] = y * cq[o] + rot * sq[o];
        }
    }
}

// ---------------------------------------------------------------------------
// Kernel 2: masked block max/avg pooling of K. One block per (b, block j).
// ---------------------------------------------------------------------------
__global__ __launch_bounds__(256) void kpool_kernel(
    const float* __restrict__ k, const int* __restrict__ cu,
    float* __restrict__ kcat)
{
    int bj = blockIdx.x;
    int b = bj / NB, j = bj % NB;
    int start = cu[b], len = cu[b + 1] - cu[b];
    int i0 = j * BLOCK;
    int cnt = len - i0; if (cnt > BLOCK) cnt = BLOCK; if (cnt < 0) cnt = 0;
    for (int c = threadIdx.x; c < KH * D; c += 256) {
        float mx = -INFINITY, sm = 0.f;
        const float* base = k + ((size_t)(start + i0)) * (KH * D) + c;
        for (int i = 0; i < cnt; ++i) {
            float v = base[(size_t)i * (KH * D)];
            mx = fmaxf(mx, v); sm += v;
        }
        float avg = (cnt > 0) ? sm / (float)cnt : 0.f;
        if (cnt == 0) mx = 0.f;
        int hh = c >> 7, d = c & 127;
        float* dst = kcat + (((size_t)b * KH + hh) * NB + j) * (2 * D);
        dst[d]     = mx;
        dst[D + d] = avg;
    }
}

// ---------------------------------------------------------------------------
// Kernel 3: K projection (64 x 256)*(256 x 128) per (b,h) with fp32 WMMA,
// fused rmsnorm + rope. Block = 128 threads (4 waves), wave owns 16 rows.
// A stride 68, B k-pair packed stride 160 (aligned b64 fragments).
// ---------------------------------------------------------------------------
__global__ __launch_bounds__(128) void kproj_kernel(
    const float* __restrict__ kcat, const float* __restrict__ wk,
    const float* __restrict__ knw, const float* __restrict__ cosk,
    const float* __restrict__ sink, float* __restrict__ kkrot)
{
    __shared__ float As[64 * 68];
    __shared__ float Bs[32 * 160];
    int bh = blockIdx.x; int b = bh / KH, h = bh % KH;
    int tid = threadIdx.x, wave = tid >> 5, lane = tid & 31;
    v8f acc[8];
    #pragma unroll
    for (int i = 0; i < 8; ++i) acc[i] = zero8();

    const float* Ag = kcat + (size_t)bh * NB * (2 * D);
    const float* Bg = wk + (size_t)h * (2 * D) * DG;

    for (int kc = 0; kc < 2 * D; kc += 64) {
        for (int i = tid; i < 64 * 16; i += 128) {
            int row = i >> 4, c4 = (i & 15) * 4;
            float4 v = *(const float4*)(Ag + (size_t)row * (2 * D) + kc + c4);
            float* dst = &As[row * 68 + c4];
            dst[0] = v.x; dst[1] = v.y; dst[2] = v.z; dst[3] = v.w;
        }
        for (int i = tid; i < 64 * 32; i += 128) {
            int row = i >> 5, c4 = (i & 31) * 4;
            float4 v = *(const float4*)(Bg + (size_t)(kc + row) * DG + c4);
            float* dst = &Bs[(row >> 1) * 160 + (row & 1)];
            dst[(c4 + 0) * 2] = v.x; dst[(c4 + 1) * 2] = v.y;
            dst[(c4 + 2) * 2] = v.z; dst[(c4 + 3) * 2] = v.w;
        }
        __syncthreads();
        int mrow = wave * 16 + (lane & 15);
        int half = lane >> 4;
        for (int kk = 0; kk < 64; kk += 4) {
            int kb = kk + half * 2;
            v2f a = *(const v2f*)&As[mrow * 68 + kb];
            int kp = (kk >> 1) + half;
            #pragma unroll
            for (int nt = 0; nt < 8; ++nt) {
                int n = nt * 16 + (lane & 15);
                v2f bb = *(const v2f*)&Bs[kp * 160 + 2 * n];
                acc[nt] = __builtin_amdgcn_wmma_f32_16x16x4_f32(
                    false, a, false, bb, (short)0, acc[nt], false, false);
            }
        }
        __syncthreads();
    }

    int g = lane >> 4;
    #pragma unroll
    for (int r = 0; r < 8; ++r) {
        int j = wave * 16 + r + g * 8;           // block index 0..63
        float p = 0.f;
        #pragma unroll
        for (int nt = 0; nt < 8; ++nt) { float v = acc[nt][r]; p += v * v; }
        p += __shfl_xor(p, 1); p += __shfl_xor(p, 2);
        p += __shfl_xor(p, 4); p += __shfl_xor(p, 8);
        float rinv = 1.0f / sqrtf(p * (1.0f / DG) + 1e-6f);
        const float* cq = cosk + ((size_t)b * NB + j) * DG;
        const float* sq = sink + ((size_t)b * NB + j) * DG;
        float* outp = kkrot + (((size_t)b * KH + h) * NB + j) * DG;
        #pragma unroll
        for (int nt = 0; nt < 8; ++nt) {
            int o = nt * 16 + (lane & 15);
            float y  = acc[nt][r]     * rinv * knw[o];
            float y2 = acc[nt ^ 4][r] * rinv * knw[o ^ 64];
            float rot = (nt < 4) ? -y2 : y2;
            outp[o] = y * cq[o] + rot * sq[o];
        }
    }
}

// ---------------------------------------------------------------------------
// Kernel 4: gate scores: (128 x 128) * (128 x 64)^T per tile with fp32 WMMA,
// scale by 1/sqrt(128), masked fill to -1e20.
// ---------------------------------------------------------------------------
__global__ __launch_bounds__(256) void attn_kernel(
    const float* __restrict__ qrot, const float* __restrict__ kkrot,
    const unsigned char* __restrict__ mask, float* __restrict__ out)
{
    __shared__ float As[128 * 68];
    __shared__ float Bs[32 * 160];
    int bh = blockIdx.y; int b = bh / KH;
    int s0 = blockIdx.x * 128;
    int tid = threadIdx.x, wave = tid >> 5, lane = tid & 31;
    v8f acc[4];
    #pragma unroll
    for (int i = 0; i < 4; ++i) acc[i] = zero8();

    const float* Ag = qrot + ((size_t)bh * MAXS + s0) * DG;
    const float* Bg = kkrot + (size_t)bh * NB * DG;

    for (int kc = 0; kc < DG; kc += 64) {
        for (int i = tid; i < 128 * 16; i += 256) {
            int row = i >> 4, c4 = (i & 15) * 4;
            float4 v = *(const float4*)(Ag + (size_t)row * DG + kc + c4);
            float* dst = &As[row * 68 + c4];
            dst[0] = v.x; dst[1] = v.y; dst[2] = v.z; dst[3] = v.w;
        }
        // transpose-stage B, packed in k-pairs: Bs[kp][2n+par] = kkrot[n][kc+k]
        for (int i = tid; i < 64 * 64; i += 256) {
            int n = i >> 6, kk = i & 63;
            Bs[(kk >> 1) * 160 + 2 * n + (kk & 1)] = Bg[(size_t)n * DG + kc + kk];
        }
        __syncthreads();
        int mrow = wave * 16 + (lane & 15);
        int half = lane >> 4;
        for (int kk = 0; kk < 64; kk += 4) {
            int kb = kk + half * 2;
            v2f a = *(const v2f*)&As[mrow * 68 + kb];
            int kp = (kk >> 1) + half;
            #pragma unroll
            for (int nt = 0; nt < 4; ++nt) {
                int n = nt * 16 + (lane & 15);
                v2f bb = *(const v2f*)&Bs[kp * 160 + 2 * n];
                acc[nt] = __builtin_amdgcn_wmma_f32_16x16x4_f32(
                    false, a, false, bb, (short)0, acc[nt], false, false);
            }
        }
        __syncthreads();
    }

    const float scale = 0.08838834764831845f;  // 1/sqrt(128)
    int g = lane >> 4;
    const unsigned char* mp = mask + (size_t)b * MAXS * NB;
    #pragma unroll
    for (int r = 0; r < 8; ++r) {
        int s = s0 + wave * 16 + r + g * 8;
        float* op = out + ((size_t)bh * MAXS + s) * NB;
        const unsigned char* mr = mp + (size_t)s * NB;
        #pragma unroll
        for (int nt = 0; nt < 4; ++nt) {
            int j = nt * 16 + (lane & 15);
            float v = acc[nt][r] * scale;
            op[j] = mr[j] ? v : -1e20f;
        }
    }
}

// ---------------------------------------------------------------------------
extern "C" void kernel_launch(void* const* d_in, const int* in_sizes, int n_in,
                              void* d_out, int out_size, void* d_ws, size_t ws_size,
                              hipStream_t stream) {
    const float* q    = (const float*)d_in[0];
    const float* k    = (const float*)d_in[1];
    const float* wq   = (const float*)d_in[2];
    const float* wk   = (const float*)d_in[3];
    const float* qnw  = (const float*)d_in[4];
    const float* knw  = (const float*)d_in[5];
    const float* cosq = (const float*)d_in[6];
    const float* sinq = (const float*)d_in[7];
    const float* cosk = (const float*)d_in[8];
    const float* sink = (const float*)d_in[9];
    const unsigned char* mask = (const unsigned char*)d_in[10];
    const int* cu     = (const int*)d_in[11];
    const long long* unpad = (const long long*)d_in[12];

    int T = in_sizes[1] / (KH * D);

    float* qrot  = (float*)d_ws;                               // 64 MB
    float* kcat  = qrot + (size_t)BSZ * KH * MAXS * DG;        // 2 MB
    float* kkrot = kcat + (size_t)BSZ * KH * NB * (2 * D);     // 1 MB
    float* out   = (float*)d_out;

    dim3 g1((T + 127) / 128, KH);
    qproj_kernel<<<g1, 256, 0, stream>>>(q, wq, qnw, cosq, sinq, unpad, qrot, T);
    kpool_kernel<<<BSZ * NB, 256, 0, stream>>>(k, cu, kcat);
    kproj_kernel<<<BSZ * KH, 128, 0, stream>>>(kcat, wk, knw, cosk, sink, kkrot);
    dim3 g4(MAXS / 128, BSZ * KH);
    attn_kernel<<<g4, 256, 0, stream>>>(qrot, kkrot, mask, out);
}